// HAMMER_87471303950520
// MI455X (gfx1250) — compile-verified
//
#include <hip/hip_runtime.h>

// Problem shape (fixed by the reference's setup_inputs)
#define B_ 2
#define V_ 5
#define C_ 32
#define H_ 256
#define W_ 320
#define D_ 4
#define HW_ (H_ * W_)

typedef __attribute__((ext_vector_type(2))) float v2f;
typedef __attribute__((ext_vector_type(8))) float v8f;

// ---------------------------------------------------------------------------
// Kernel 1: one wave. Builds comb[b][v] = [K3*E[:3,:4]; 0 0 0 1], inverts the
// affine reference matrix, then computes proj[b][v] = comb[b][v] * inv(ref)
// for v=1..4 with V_WMMA_F32_16X16X4_F32 (four 4x4 products per WMMA:
// A = 16x4 stack of the four source matrices, B = 4x16 inv(ref) replicated
// across column blocks, so D rows 4i..4i+3 / cols 0..3 hold product i).
// Emits rot (row-major 3x3) + trans (3) per (b, v-1) into ws (12 floats each).
// ---------------------------------------------------------------------------
__global__ void hammer_setup_mats(const float* __restrict__ pm,
                                  float* __restrict__ ws) {
  __shared__ float comb[B_][V_][16];
  __shared__ float invref[B_][16];
  const int lane = threadIdx.x;  // 32 threads, wave32

  if (lane < B_ * V_) {
    const int b = lane / V_, v = lane % V_;
    const float* E = pm + ((b * V_ + v) * 2 + 0) * 16;
    const float* K = pm + ((b * V_ + v) * 2 + 1) * 16;
    for (int r = 0; r < 3; ++r)
      for (int c = 0; c < 4; ++c)
        comb[b][v][r * 4 + c] =
            K[r * 4 + 0] * E[0 + c] + K[r * 4 + 1] * E[4 + c] + K[r * 4 + 2] * E[8 + c];
    for (int c = 0; c < 4; ++c) comb[b][v][12 + c] = E[12 + c];  // [0,0,0,1]
  }
  __syncthreads();

  if (lane < B_) {
    const int b = lane;
    const float* Cm = comb[b][0];
    const float m00 = Cm[0], m01 = Cm[1], m02 = Cm[2], t0 = Cm[3];
    const float m10 = Cm[4], m11 = Cm[5], m12 = Cm[6], t1 = Cm[7];
    const float m20 = Cm[8], m21 = Cm[9], m22 = Cm[10], t2 = Cm[11];
    const float c00 = (m11 * m22 - m12 * m21);
    const float c01 = -(m10 * m22 - m12 * m20);
    const float c02 = (m10 * m21 - m11 * m20);
    const float id = 1.0f / (m00 * c00 + m01 * c01 + m02 * c02);
    const float i00 = c00 * id;
    const float i01 = -(m01 * m22 - m02 * m21) * id;
    const float i02 = (m01 * m12 - m02 * m11) * id;
    const float i10 = c01 * id;
    const float i11 = (m00 * m22 - m02 * m20) * id;
    const float i12 = -(m00 * m12 - m02 * m10) * id;
    const float i20 = c02 * id;
    const float i21 = -(m00 * m21 - m01 * m20) * id;
    const float i22 = (m00 * m11 - m01 * m10) * id;
    invref[b][0] = i00; invref[b][1] = i01; invref[b][2]  = i02;
    invref[b][3] = -(i00 * t0 + i01 * t1 + i02 * t2);
    invref[b][4] = i10; invref[b][5] = i11; invref[b][6]  = i12;
    invref[b][7] = -(i10 * t0 + i11 * t1 + i12 * t2);
    invref[b][8] = i20; invref[b][9] = i21; invref[b][10] = i22;
    invref[b][11] = -(i20 * t0 + i21 * t1 + i22 * t2);
    invref[b][12] = 0.f; invref[b][13] = 0.f; invref[b][14] = 0.f; invref[b][15] = 1.f;
  }
  __syncthreads();

  const int m = lane & 15;    // A row / B-D column
  const int kh = lane >> 4;   // half-wave select (K/row split per ISA layout)

  for (int b = 0; b < B_; ++b) {
    // A[m][k]: row m -> source matrix (m>>2)+1, its row m&3.
    const int vi = (m >> 2) + 1, r = m & 3;
    v2f a;
    a.x = comb[b][vi][r * 4 + 2 * kh + 0];
    a.y = comb[b][vi][r * 4 + 2 * kh + 1];
    // B[k][n] = invref[k][n&3] (replicated column blocks).
    v2f bm;
    bm.x = invref[b][(2 * kh + 0) * 4 + (m & 3)];
    bm.y = invref[b][(2 * kh + 1) * 4 + (m & 3)];
    v8f cz = {0.f, 0.f, 0.f, 0.f, 0.f, 0.f, 0.f, 0.f};
    v8f dd = __builtin_amdgcn_wmma_f32_16x16x4_f32(
        false, a, false, bm, (short)0, cz, false, false);
    // D layout: lane holds column n=m, rows j + 8*kh in dd[j]. Read col block 0.
    if (m < 4) {
#pragma unroll
      for (int j = 0; j < 8; ++j) {
        const int R = j + 8 * kh;          // global row
        const int i = R >> 2, rr = R & 3;  // product i (view i+1), its row rr
        if (rr < 3) {
          const float val = dd[j];
          float* o = ws + (b * (V_ - 1) + i) * 12;
          if (m < 3) o[rr * 3 + m] = val;  // rot, row-major
          else       o[9 + rr]     = val;  // trans
        }
      }
    }
  }
}

// ---------------------------------------------------------------------------
// Kernel 2: zero the cv_mask region (d_out is poisoned; atomics accumulate).
// ---------------------------------------------------------------------------
__global__ void hammer_zero_mask(float* __restrict__ mask) {
  const int i = blockIdx.x * 256 + threadIdx.x;
  if (i < B_ * HW_) mask[i] = 0.0f;
}

// ---------------------------------------------------------------------------
// Kernel 3: main warp + variance. One thread per (b,d,h,w); consecutive
// threads walk w (coalesced depth load, near-coalesced gathers, coalesced
// strided variance stores). 32 sum + 32 sumsq accumulators live in VGPRs.
// Bilinear with zero padding == per-tap weight zeroing + clamped indices.
// ---------------------------------------------------------------------------
__global__ __launch_bounds__(256) void hammer_warp_var(
    const float* __restrict__ feats, const float* __restrict__ depth,
    const float* __restrict__ ws, float* __restrict__ outVar,
    float* __restrict__ outMask) {
  const int idx = blockIdx.x * 256 + threadIdx.x;
  const int w = idx % W_;
  int t = idx / W_;
  const int h = t % H_; t /= H_;
  const int d = t % D_;           // uniform per block (W*H % 256 == 0)
  const int b = t / D_;           // uniform per block
  const int p = h * W_ + w;

  const float dep = depth[(b * D_ + d) * HW_ + p];

  float s[C_], q[C_];
  const float* ref = feats + (size_t)(b * V_) * C_ * HW_ + p;
#pragma unroll
  for (int c = 0; c < C_; ++c) {
    const float f = ref[c * HW_];
    s[c] = f;
    q[c] = f * f;
  }

  const float xs = (float)w, ys = (float)h;
  int cnt = 0;

  for (int v = 1; v < V_; ++v) {
    const float* M = ws + (b * (V_ - 1) + (v - 1)) * 12;
    const float px = (M[0] * xs + M[1] * ys + M[2]) * dep + M[9];
    const float py = (M[3] * xs + M[4] * ys + M[5]) * dep + M[10];
    const float pz = (M[6] * xs + M[7] * ys + M[8]) * dep + M[11];
    const float x = px / pz;   // grid normalize/denormalize cancels exactly
    const float y = py / pz;

    const float x0f = floorf(x), y0f = floorf(y);
    const float fx = x - x0f, fy = y - y0f;
    const float gx = 1.0f - fx, gy = 1.0f - fy;
    const bool vx0 = (x0f >= 0.0f) && (x0f <= (float)(W_ - 1));
    const bool vx1 = (x0f >= -1.0f) && (x0f <= (float)(W_ - 2));
    const bool vy0 = (y0f >= 0.0f) && (y0f <= (float)(H_ - 1));
    const bool vy1 = (y0f >= -1.0f) && (y0f <= (float)(H_ - 2));
    const float w00 = gx * gy * (float)(vx0 && vy0);
    const float w10 = fx * gy * (float)(vx1 && vy0);
    const float w01 = gx * fy * (float)(vx0 && vy1);
    const float w11 = fx * fy * (float)(vx1 && vy1);
    const int ix0 = (int)fminf(fmaxf(x0f, 0.0f), (float)(W_ - 1));
    const int ix1 = (int)fminf(fmaxf(x0f + 1.0f, 0.0f), (float)(W_ - 1));
    const int iy0 = (int)fminf(fmaxf(y0f, 0.0f), (float)(H_ - 1));
    const int iy1 = (int)fminf(fmaxf(y0f + 1.0f, 0.0f), (float)(H_ - 1));
    const int o00 = iy0 * W_ + ix0, o10 = iy0 * W_ + ix1;
    const int o01 = iy1 * W_ + ix0, o11 = iy1 * W_ + ix1;

    const float* src = feats + (size_t)(b * V_ + v) * C_ * HW_;
    bool any = false;
#pragma unroll
    for (int c = 0; c < C_; ++c) {
      const float* sc = src + c * HW_;
      const float val =
          w00 * sc[o00] + w10 * sc[o10] + w01 * sc[o01] + w11 * sc[o11];
      s[c] += val;
      q[c] += val * val;
      any = any || (val != 0.0f);
    }
    cnt += any ? 1 : 0;
  }

  // mask: exact small-integer float adds -> order independent, deterministic.
  atomicAdd(&outMask[b * HW_ + p], (float)cnt);

  const float invV = 1.0f / (float)V_;
  float* ov = outVar + ((size_t)b * C_ * D_ + d) * HW_ + p;
#pragma unroll
  for (int c = 0; c < C_; ++c) {
    const float mean = s[c] * invV;
    ov[(size_t)c * (D_ * HW_)] = q[c] * invV - mean * mean;
  }
}

// ---------------------------------------------------------------------------
extern "C" void kernel_launch(void* const* d_in, const int* in_sizes, int n_in,
                              void* d_out, int out_size, void* d_ws,
                              size_t ws_size, hipStream_t stream) {
  const float* feats = (const float*)d_in[0];  // [B,V,C,H,W]
  const float* pm    = (const float*)d_in[1];  // [B,V,2,4,4]
  const float* depth = (const float*)d_in[2];  // [B,D,H,W]
  float* outVar  = (float*)d_out;                           // [B,C,D,H,W]
  float* outMask = outVar + (size_t)B_ * C_ * D_ * HW_;     // [B,H,W]
  float* ws = (float*)d_ws;  // 96 floats: per (b, v-1) rot(9)+trans(3)

  hammer_setup_mats<<<1, 32, 0, stream>>>(pm, ws);
  hammer_zero_mask<<<(B_ * HW_ + 255) / 256, 256, 0, stream>>>(outMask);
  const int total = B_ * D_ * HW_;  // 655,360 -> 2,560 blocks
  hammer_warp_var<<<total / 256, 256, 0, stream>>>(feats, depth, ws, outVar,
                                                   outMask);
}